// YOLOLayer_32117765439761
// MI455X (gfx1250) — compile-verified
//
#include <hip/hip_runtime.h>
#include <hip/hip_bf16.h>
#include <stdint.h>

// ---------------------------------------------------------------------------
// YOLOv4 head post-processing for MI455X (gfx1250).
//
// Memory-bound elementwise + CHW->HWC transpose: ~187.5 MB traffic,
// roofline ~8.0 us @ 23.3 TB/s. Strategy:
//   * One workgroup per (batch, anchor, row): tile = 85 channels x 76 px,
//     row stride 5776 floats in global memory.
//   * Tensor Data Mover (TENSOR_LOAD_TO_LDS) DMAs the whole strided tile
//     into LDS with ONE instruction (wave 0 only), tracked via TENSORcnt.
//   * One transcendental per input element (sigmoid/exp applied once in a
//     transform pass), results stored to an odd-stride (77) LDS buffer so
//     the per-pixel channel gather is bank-conflict free.
//   * Output pass writes the 76x84 HWC tile as float4 (global_store_b128),
//     fully coalesced; the j<4 float4 group is exactly one box.
// ---------------------------------------------------------------------------

typedef unsigned int u32x4 __attribute__((ext_vector_type(4)));
typedef int          i32x4 __attribute__((ext_vector_type(4)));
typedef int          i32x8 __attribute__((ext_vector_type(8)));

#define YL_A     3
#define YL_NC    80
#define YL_CH    85          // 5 + NC
#define YL_H     76
#define YL_W     76
#define YL_HW    (YL_H * YL_W)   // 5776
#define YL_OUTC  84          // 4 + NC
#define YL_PW    (YL_W + 1)  // padded LDS row stride (odd -> no bank conflicts)
#define YL_G     (YL_OUTC / 4)   // 21 float4 groups per pixel

#define LOG2E 1.44269504088896341f

__device__ __forceinline__ float fast_sigmoid(float v) {
    // 1 / (1 + exp(-v)) via exp2 + rcp (one trans pair per element)
    float e = __builtin_amdgcn_exp2f(v * (-LOG2E));
    return __builtin_amdgcn_rcpf(1.0f + e);
}

__global__ void yolo_head_kernel(const float* __restrict__ x,
                                 float* __restrict__ out) {
    extern __shared__ float smem[];
    float* raw = smem;                    // [85][76] raw tile (TDM target, LDS offset 0)
    float* tp  = smem + YL_CH * YL_W;     // [85][77] transformed, padded stride

    const int bid = blockIdx.x;           // ((b*3)+a)*76 + h
    const int h   = bid % YL_H;
    const int ba  = bid / YL_H;           // b*3 + a
    const int a   = ba % YL_A;
    const int b   = ba / YL_A;

    // Input tile start: channel 0 of this (b,a) slab, row h.
    const size_t inBase = (size_t)ba * YL_CH * YL_HW + (size_t)h * YL_W;

    // ---------------- TDM descriptor (D#) ----------------
    const unsigned long long gaddr = (unsigned long long)(uintptr_t)(x + inBase);

    u32x4 g0;
    g0[0] = 1u;                                        // count=1, user mode, no gather
    g0[1] = 0u;                                        // lds_addr = 0 (dynamic LDS base)
    g0[2] = (unsigned)(gaddr & 0xFFFFFFFFull);         // global_addr[31:0]
    g0[3] = (unsigned)((gaddr >> 32) & 0x01FFFFFFull)  // global_addr[56:32]
          | (2u << 30);                                // type = 2 ("image")

    const unsigned td0 = (unsigned)(YL_HW - h * YL_W); // remaining dim0 extent (OOB guard)
    i32x8 g1;
    g1[0] = (int)(2u << 16);                           // data_size = 2 (4 bytes)
    g1[1] = (int)((td0 & 0xFFFFu) << 16);              // tensor_dim0[15:0]
    g1[2] = (int)(((td0 >> 16) & 0xFFFFu)              // tensor_dim0[31:16]
          | ((unsigned)YL_CH << 16));                  // tensor_dim1[15:0] = 85
    g1[3] = (int)((unsigned)YL_W << 16);               // tile_dim0 = 76
    g1[4] = (int)(unsigned)YL_CH;                      // tile_dim1 = 85, tile_dim2 = 0
    g1[5] = (int)(unsigned)YL_HW;                      // tensor_dim0_stride = 5776
    g1[6] = 0;                                         // stride hi / dim1_stride lo
    g1[7] = 0;
    i32x4 gz  = {0, 0, 0, 0};                          // groups 2/3 unused (2-D tensor)
    i32x8 gz8 = {0, 0, 0, 0, 0, 0, 0, 0};              // extra group (unused)

    if (threadIdx.x < 32) {                            // wave 0 only: one DMA per block
        __builtin_amdgcn_tensor_load_to_lds(g0, g1, gz, gz, gz8, 0);
        __builtin_amdgcn_s_wait_tensorcnt(0);
    }
    __syncthreads();

    // ---------------- Transform pass: one trans op per element ----------------
    // c==0,1: sigmoid(v)*1.2 - 0.1   (scale_xy)
    // c==2,3: exp(v)
    // c>=4 : sigmoid(v)              (det + cls)
    for (int idx = threadIdx.x; idx < YL_CH * YL_W; idx += blockDim.x) {
        const int c = idx / YL_W;
        const int s = idx - c * YL_W;
        const float v = raw[idx];
        float r;
        if (c == 2 || c == 3) {
            r = __builtin_amdgcn_exp2f(v * LOG2E);
        } else {
            r = fast_sigmoid(v);
            if (c < 2) r = r * 1.2f - 0.1f;
        }
        tp[c * YL_PW + s] = r;
    }
    __syncthreads();

    // ---------------- Output pass: coalesced float4 HWC writes ----------------
    // anchors / stride: [12,16],[19,36],[40,28] / 8
    const float anchx[YL_A] = {1.5f, 2.375f, 5.0f};
    const float anchy[YL_A] = {2.0f, 4.5f, 3.5f};
    const float ax = anchx[a], ay = anchy[a];
    const float invW = 1.0f / (float)YL_W;
    const float gy = (float)h;

    const size_t outBase =
        ((size_t)b * (YL_A * YL_HW) + (size_t)a * YL_HW + (size_t)h * YL_W) * YL_OUTC;
    float4* out4 = (float4*)(out + outBase);           // 336-byte pixel rows: 16B aligned

    const int NG = YL_W * YL_G;                        // 1596 float4 groups per block
    for (int g = threadIdx.x; g < NG; g += blockDim.x) {
        const int s  = g / YL_G;                       // pixel (w)
        const int jb = (g - s * YL_G) * 4;             // base output channel (mult. of 4)
        float4 r;
        if (jb == 0) {
            // whole box in one group: (bx1, by1, bx2, by2)
            const float p0 = tp[0 * YL_PW + s];
            const float p1 = tp[1 * YL_PW + s];
            const float p2 = tp[2 * YL_PW + s];
            const float p3 = tp[3 * YL_PW + s];
            const float bx = (p0 + (float)s) * invW;
            const float by = (p1 + gy) * invW;
            const float bw = p2 * ax * invW;
            const float bh = p3 * ay * invW;
            r.x = bx - bw * 0.5f;
            r.y = by - bh * 0.5f;
            r.z = r.x + bw;
            r.w = r.y + bh;
        } else {
            // four conf products: cls channel = j + 1 = jb+1 .. jb+4
            const float det = tp[4 * YL_PW + s];
            r.x = tp[(jb + 1) * YL_PW + s] * det;
            r.y = tp[(jb + 2) * YL_PW + s] * det;
            r.z = tp[(jb + 3) * YL_PW + s] * det;
            r.w = tp[(jb + 4) * YL_PW + s] * det;
        }
        out4[g] = r;
    }
}

extern "C" void kernel_launch(void* const* d_in, const int* in_sizes, int n_in,
                              void* d_out, int out_size, void* d_ws, size_t ws_size,
                              hipStream_t stream) {
    (void)in_sizes; (void)n_in; (void)out_size; (void)d_ws; (void)ws_size;
    const float* x = (const float*)d_in[0];
    float* out = (float*)d_out;

    const int blocks = 16 * YL_A * YL_H;               // 3648
    const size_t smem = (size_t)(YL_CH * YL_W + YL_CH * YL_PW) * sizeof(float); // ~52 KB
    yolo_head_kernel<<<blocks, 256, smem, stream>>>(x, out);
}